// My_loss_27676769255433
// MI455X (gfx1250) — compile-verified
//
#include <hip/hip_runtime.h>

typedef __attribute__((ext_vector_type(2))) float v2f;
typedef __attribute__((ext_vector_type(4))) float v4f;
typedef __attribute__((ext_vector_type(8))) float v8f;
typedef __attribute__((ext_vector_type(2))) int   v2i;
typedef __attribute__((ext_vector_type(4))) int   v4i;

#define BLOCK 256
#define WAVES_PER_BLOCK 8
#define TILES 4                 // 4 WMMA column tiles -> 64 contiguous columns per block
#define COLS_PER_BLOCK (TILES * 16)
#define KCHUNKS 4               // K split across blockIdx.y

// Convert int64 labels -> packed int32 (enables b64 class loads in hot loop).
__global__ void prep_labels_kernel(const long long* __restrict__ y1,
                                   const long long* __restrict__ y2,
                                   int N1, int N2,
                                   int* __restrict__ c1, int* __restrict__ c2) {
    int i = blockIdx.x * blockDim.x + threadIdx.x;
    if (i < N1) c1[i] = (int)y1[i];
    if (i < N2) c2[i] = (int)y2[i];
}

// For a 64-column strip and K-chunk of `mat` [K x N]:
//   masked = sum of mat[k,n] * (rowCls[k] == colCls[n]);  total = sum of mat[k,n]
// via D = onehot(rowCls) x (mat tile) on V_WMMA_F32_16X16X4_F32.
// Physical column of (tile w, N-index nlo) is cbase + 4*nlo + w, so each lane's
// b128 load feeds all 4 accumulator tiles with fully contiguous 256B/half-wave.
__global__ void masked_colsum_kernel(const float* __restrict__ mat,
                                     const int* __restrict__ rowCls,
                                     const int* __restrict__ colCls,
                                     int N, int kChunk,
                                     float* __restrict__ partials /* [grid][2] */) {
    __shared__ float red[BLOCK][2];

    const int lane  = threadIdx.x & 31;
    const int wave  = threadIdx.x >> 5;
    const int half  = lane >> 4;           // 0: K rows 0/1, 1: K rows 2/3
    const int nlo   = lane & 15;
    const int cbase = blockIdx.x * COLS_PER_BLOCK;
    const int myM   = nlo;                 // A-matrix row (class id) for this lane

    // UNIFORM trip count (SGPR): scalar loop control, no EXEC masking near WMMA.
    const int iters = kChunk / (WAVES_PER_BLOCK * 4);

    // Lane's start row and bump pointers (strength-reduced addressing).
    const int kStart = blockIdx.y * kChunk + wave * (kChunk / WAVES_PER_BLOCK) + half * 2;
    const float* p  = mat + (size_t)kStart * (size_t)N + (size_t)cbase + (size_t)(4 * nlo);
    const int*   cp = rowCls + kStart;
    const size_t rowStep = (size_t)N;      // floats per row

    v8f acc0 = {}, acc1 = {}, acc2 = {}, acc3 = {};
#pragma unroll 2
    for (int it = 0; it < iters; ++it) {
        // Two b128 row fragments: rows kStart+4it and +1, cols cbase+4*nlo+0..3.
        const v4f r0 = *(const v4f*)(p);
        const v4f r1 = *(const v4f*)(p + rowStep);
        // One b64 class load for the two rows; one-hot A shared by all 4 tiles.
        const v2i cc = *(const v2i*)(cp);
        p  += 4 * rowStep;
        cp += 4;

        v2f a;
        a.x = (cc.x == myM) ? 1.0f : 0.0f;
        a.y = (cc.y == myM) ? 1.0f : 0.0f;

        v2f b;
        b.x = r0.x; b.y = r1.x;
        acc0 = __builtin_amdgcn_wmma_f32_16x16x4_f32(false, a, false, b, (short)0, acc0, false, false);
        b.x = r0.y; b.y = r1.y;
        acc1 = __builtin_amdgcn_wmma_f32_16x16x4_f32(false, a, false, b, (short)0, acc1, false, false);
        b.x = r0.z; b.y = r1.z;
        acc2 = __builtin_amdgcn_wmma_f32_16x16x4_f32(false, a, false, b, (short)0, acc2, false, false);
        b.x = r0.w; b.y = r1.w;
        acc3 = __builtin_amdgcn_wmma_f32_16x16x4_f32(false, a, false, b, (short)0, acc3, false, false);
    }

    // Column classes for this lane's 4 physical columns (one b128).
    const v4i cw = *(const v4i*)(colCls + cbase + 4 * nlo);

    // C/D layout: accW[v] = D_w[v + half*8, col = cbase + 4*nlo + w]
    float masked = 0.0f, total = 0.0f;
#pragma unroll
    for (int v = 0; v < 8; ++v) {
        const int m = v + half * 8;
        const float d0 = acc0[v], d1 = acc1[v], d2 = acc2[v], d3 = acc3[v];
        total += d0 + d1 + d2 + d3;
        masked += (m == cw.x) ? d0 : 0.0f;
        masked += (m == cw.y) ? d1 : 0.0f;
        masked += (m == cw.z) ? d2 : 0.0f;
        masked += (m == cw.w) ? d3 : 0.0f;
    }

    red[threadIdx.x][0] = masked;
    red[threadIdx.x][1] = total;
    __syncthreads();
    if (threadIdx.x == 0) {
        float sm = 0.0f, st = 0.0f;
        for (int i = 0; i < BLOCK; ++i) { sm += red[i][0]; st += red[i][1]; }
        const size_t pidx = (size_t)blockIdx.y * gridDim.x + blockIdx.x;
        partials[2 * pidx + 0] = sm;
        partials[2 * pidx + 1] = st;
    }
}

// Single block: label histograms (integer LDS atomics, order-independent),
// fixed-order partial sums, final fp32 scalar (matches jnp.float32 math).
__global__ void finalize_kernel(const int* __restrict__ c1,
                                const int* __restrict__ c2,
                                int N1, int N2,
                                const float* __restrict__ part_f2, int nblk2,
                                const float* __restrict__ part_f,  int nblk1,
                                float* __restrict__ out) {
    __shared__ int h1[16];
    __shared__ int h2[16];
    if (threadIdx.x < 16) { h1[threadIdx.x] = 0; h2[threadIdx.x] = 0; }
    __syncthreads();
    for (int i = threadIdx.x; i < N1; i += blockDim.x) atomicAdd(&h1[c1[i] & 15], 1);
    for (int i = threadIdx.x; i < N2; i += blockDim.x) atomicAdd(&h2[c2[i] & 15], 1);
    __syncthreads();
    if (threadIdx.x == 0) {
        float m1 = 0.0f, sumF2 = 0.0f;
        for (int i = 0; i < nblk2; ++i) { m1 += part_f2[2 * i]; sumF2 += part_f2[2 * i + 1]; }
        float m2 = 0.0f, sumF = 0.0f;
        for (int i = 0; i < nblk1; ++i) { m2 += part_f[2 * i]; sumF += part_f[2 * i + 1]; }

        float same22 = 0.0f, same12 = 0.0f;
        for (int c = 0; c < 16; ++c) {
            same22 += (float)h2[c] * (float)h2[c];
            same12 += (float)h1[c] * (float)h2[c];
        }
        const float n2f = (float)N2;
        const float same1      = same22 - n2f;
        const float different1 = n2f * n2f - same22;        // N2*N2 - (same1 + N2)
        const float same2      = same12;
        const float different2 = (float)N1 * n2f - same2;
        const float n1 = sumF2 - m1;
        const float n2 = sumF - m2;
        out[0] = (m1 / same1 + m2 / same2) / (n1 / different1 + n2 / different2);
    }
}

extern "C" void kernel_launch(void* const* d_in, const int* in_sizes, int n_in,
                              void* d_out, int out_size, void* d_ws, size_t ws_size,
                              hipStream_t stream) {
    const float* f  = (const float*)d_in[0];         // [N1, N2] fp32
    const float* f2 = (const float*)d_in[1];         // [N2, N2] fp32
    const long long* y1 = (const long long*)d_in[2]; // [N1] int64
    const long long* y2 = (const long long*)d_in[3]; // [N2] int64

    const int N1 = in_sizes[2];
    const int N2 = in_sizes[3];

    // Workspace layout: [c1: N1 ints][c2: N2 ints][part_f2][part_f]
    int* c1 = (int*)d_ws;
    int* c2 = c1 + N1;
    const int gx = N2 / COLS_PER_BLOCK;              // 128 column strips
    const int pb = gx * KCHUNKS;                     // partial blocks per matrix
    float* part_f2 = (float*)(c2 + N2);
    float* part_f  = part_f2 + 2 * (size_t)pb;

    const int nmax = (N1 > N2) ? N1 : N2;
    prep_labels_kernel<<<(nmax + BLOCK - 1) / BLOCK, BLOCK, 0, stream>>>(y1, y2, N1, N2, c1, c2);

    dim3 grid(gx, KCHUNKS);
    // f2: rows labeled y2, cols labeled y2, K = N2
    masked_colsum_kernel<<<grid, BLOCK, 0, stream>>>(f2, c2, c2, N2, N2 / KCHUNKS, part_f2);
    // f: rows labeled y1, cols labeled y2, K = N1
    masked_colsum_kernel<<<grid, BLOCK, 0, stream>>>(f,  c1, c2, N1, N1 / KCHUNKS, part_f);

    finalize_kernel<<<1, BLOCK, 0, stream>>>(c1, c2, N1, N2,
                                             part_f2, pb, part_f, pb,
                                             (float*)d_out);
}